// MTAD_MODEL_85607288143876
// MI455X (gfx1250) — compile-verified
//
#include <hip/hip_runtime.h>
#include <math.h>

typedef __attribute__((ext_vector_type(2))) float v2f;
typedef __attribute__((ext_vector_type(8))) float v8f;

#define B_   32
#define W_   100
#define F_   128
#define KS_  7
#define EMB_ 8
#define ALPHA_ 0.2f

// ---------------------------------------------------------------------------
// 1) positional encoding: xpe = x + sin(w*div)+cos(w*div), div=exp(f*-ln(1e4)/F)
// ---------------------------------------------------------------------------
__global__ void pe_kernel(const float* __restrict__ x, float* __restrict__ xpe, int n) {
    int idx = blockIdx.x * blockDim.x + threadIdx.x;
    if (idx >= n) return;
    int rem = idx % (W_ * F_);
    int w = rem / F_;
    int f = rem % F_;
    float div = __expf((float)f * (-9.210340371976184f / (float)F_));
    float arg = (float)w * div;
    xpe[idx] = x[idx] + __sinf(arg) + __cosf(arg);
}

// ---------------------------------------------------------------------------
// 2) Conv1d as WMMA: y[bt, fo] = relu( sum_{tap,fi} xpe[b, t+tap-3, fi]*w[fo,fi,tap] + cb[fo] )
//    M = B*W = 3200 (exactly 200 tiles), N = F = 128 (8 tiles), K = 128 per tap.
// ---------------------------------------------------------------------------
__global__ void conv_wmma_kernel(const float* __restrict__ xpe,
                                 const float* __restrict__ w,
                                 const float* __restrict__ cb,
                                 float* __restrict__ y) {
    int wave = threadIdx.x >> 5;
    int lane = threadIdx.x & 31;
    int half = lane >> 4;
    int l15  = lane & 15;

    int tile = blockIdx.x * 8 + wave;      // 1600 tiles total, exact
    int tm = (tile >> 3) * 16;
    int tn = (tile & 7) * 16;

    int mrow = tm + l15;                   // flattened (b,t), < 3200
    int bI = mrow / W_;
    int t  = mrow % W_;
    int ncol = tn + l15;                   // fo, < 128

    v8f acc = {};
    for (int tap = 0; tap < KS_; ++tap) {
        int tp = t + tap - 3;
        bool rOK = (tp >= 0) && (tp < W_);
        for (int k = 0; k < F_; k += 4) {
            int k0 = k + half * 2;
            v2f a, b;
            a[0] = rOK ? xpe[(bI * W_ + tp) * F_ + k0]     : 0.0f;
            a[1] = rOK ? xpe[(bI * W_ + tp) * F_ + k0 + 1] : 0.0f;
            b[0] = w[ncol * (F_ * KS_) + k0 * KS_ + tap];
            b[1] = w[ncol * (F_ * KS_) + (k0 + 1) * KS_ + tap];
            acc = __builtin_amdgcn_wmma_f32_16x16x4_f32(
                false, a, false, b, (short)0, acc, false, false);
        }
    }
    float bias = cb[ncol];
    for (int i = 0; i < 8; ++i) {
        int cm = tm + i + half * 8;
        float v = fmaxf(acc[i] + bias, 0.0f);
        y[cm * F_ + ncol] = v;
    }
}

// ---------------------------------------------------------------------------
// 3) Generic strided/batched WMMA f32 GEMM:
//    C[b][m][n] = act( sum_k A[b](m,k) * B[b](k,n) + bias )
//    A element: A + b*bsA + m*rsA + k*csA    (strides in elements)
//    B element: B + b*bsB + k*rsB + n*csB
//    biasMode: 0=none, 1=per-col bias[n], 2=scalar bias[0]
//    act: 0=none, 1=relu, 2=tanh
// ---------------------------------------------------------------------------
__global__ void gemm_wmma_kernel(const float* __restrict__ A,
                                 const float* __restrict__ Bm,
                                 const float* __restrict__ bias,
                                 float* __restrict__ C,
                                 int M, int N, int K,
                                 long bsA, int rsA, int csA,
                                 long bsB, int rsB, int csB,
                                 long bsC, int ldc,
                                 int biasMode, int act) {
    int wave = threadIdx.x >> 5;
    int lane = threadIdx.x & 31;
    int half = lane >> 4;
    int l15  = lane & 15;

    int tilesN = (N + 15) >> 4;
    int tilesM = (M + 15) >> 4;
    int tile = blockIdx.x * (blockDim.x >> 5) + wave;
    if (tile >= tilesM * tilesN) return;   // uniform per wave

    int tm = (tile / tilesN) * 16;
    int tn = (tile % tilesN) * 16;

    const float* Ab = A  + (long)blockIdx.y * bsA;
    const float* Bb = Bm + (long)blockIdx.y * bsB;
    float*       Cb = C  + (long)blockIdx.y * bsC;

    int mrow = tm + l15;
    int ncol = tn + l15;
    bool mOK = mrow < M;
    bool nOK = ncol < N;

    v8f acc = {};
    for (int k = 0; k < K; k += 4) {
        int k0 = k + half * 2;
        int k1 = k0 + 1;
        v2f a, b;
        a[0] = (mOK && k0 < K) ? Ab[(long)mrow * rsA + (long)k0 * csA] : 0.0f;
        a[1] = (mOK && k1 < K) ? Ab[(long)mrow * rsA + (long)k1 * csA] : 0.0f;
        b[0] = (nOK && k0 < K) ? Bb[(long)k0 * rsB + (long)ncol * csB] : 0.0f;
        b[1] = (nOK && k1 < K) ? Bb[(long)k1 * rsB + (long)ncol * csB] : 0.0f;
        acc = __builtin_amdgcn_wmma_f32_16x16x4_f32(
            false, a, false, b, (short)0, acc, false, false);
    }

    for (int i = 0; i < 8; ++i) {
        int cm = tm + i + half * 8;
        if (cm < M && nOK) {
            float v = acc[i];
            if (biasMode == 1)      v += bias[ncol];
            else if (biasMode == 2) v += bias[0];
            if (act == 1)      v = fmaxf(v, 0.0f);
            else if (act == 2) v = tanhf(v);
            Cb[(long)cm * ldc + ncol] = v;
        }
    }
}

// ---------------------------------------------------------------------------
// 4) GATv2 edge scores:
//    e[b,i,j] = sum_e a[e]*lrelu(left[b,i,e] + right[b,j,e] + lin_b[e]) + bias[i,j]
//    one block per (b,i); 128 threads over j; left-row + a cached in LDS.
// ---------------------------------------------------------------------------
__global__ void gat_edge_kernel(const float* __restrict__ left,
                                const float* __restrict__ right,
                                const float* __restrict__ av,
                                const float* __restrict__ lin_b,
                                const float* __restrict__ bias,
                                float* __restrict__ e,
                                int ni, int nj, int E) {
    __shared__ float sL[256];
    __shared__ float sA[256];
    int b = blockIdx.x / ni;
    int i = blockIdx.x % ni;
    int tid = threadIdx.x;
    for (int idx = tid; idx < E; idx += blockDim.x) {
        sL[idx] = left[((long)(b * ni + i)) * E + idx] + lin_b[idx];
        sA[idx] = av[idx];
    }
    __syncthreads();
    int j = tid;
    if (j < nj) {
        const float* rr = right + ((long)(b * nj + j)) * E;
        float acc = 0.0f;
        for (int k = 0; k < E; ++k) {
            float s = sL[k] + rr[k];
            float l = (s >= 0.0f) ? s : (ALPHA_ * s);
            acc += sA[k] * l;
        }
        e[((long)(b * ni + i)) * nj + j] = acc + bias[i * nj + j];
    }
}

// ---------------------------------------------------------------------------
// 5) row softmax (L <= 128), in place; one block per row, 128 threads
// ---------------------------------------------------------------------------
__global__ void softmax_rows_kernel(float* __restrict__ e, int L) {
    __shared__ float red[128];
    long row = blockIdx.x;
    int tid = threadIdx.x;
    float v = (tid < L) ? e[row * L + tid] : -3.0e38f;
    red[tid] = v;
    __syncthreads();
    for (int s = 64; s > 0; s >>= 1) {
        if (tid < s) red[tid] = fmaxf(red[tid], red[tid + s]);
        __syncthreads();
    }
    float mx = red[0];
    __syncthreads();
    float ex = (tid < L) ? __expf(v - mx) : 0.0f;
    red[tid] = ex;
    __syncthreads();
    for (int s = 64; s > 0; s >>= 1) {
        if (tid < s) red[tid] += red[tid + s];
        __syncthreads();
    }
    float inv = 1.0f / red[0];
    if (tid < L) e[row * L + tid] = ex * inv;
}

// ---------------------------------------------------------------------------
// 6) cosine similarity matrix from embedding table (F x EMB)
// ---------------------------------------------------------------------------
__global__ void cos_kernel(const float* __restrict__ emb, float* __restrict__ cosm) {
    int idx = blockIdx.x * blockDim.x + threadIdx.x;
    if (idx >= F_ * F_) return;
    int f = idx >> 7;
    int g = idx & 127;
    float d = 0.0f, nf = 0.0f, ng = 0.0f;
    for (int k = 0; k < EMB_; ++k) {
        float ef = emb[f * EMB_ + k];
        float eg = emb[g * EMB_ + k];
        d  += ef * eg;
        nf += ef * ef;
        ng += eg * eg;
    }
    cosm[idx] = d * __frsqrt_rn(nf) * __frsqrt_rn(ng);
}

// ---------------------------------------------------------------------------
// 7) z[b,g] = sum_w fc_w[w] * fout[b, g, w]   (fout is (B, F, W))
// ---------------------------------------------------------------------------
__global__ void z_kernel(const float* __restrict__ fout, const float* __restrict__ fcw,
                         float* __restrict__ z) {
    int idx = blockIdx.x * blockDim.x + threadIdx.x;
    if (idx >= B_ * F_) return;
    const float* row = fout + (long)idx * W_;
    float acc = 0.0f;
    for (int w = 0; w < W_; ++w) acc += fcw[w] * row[w];
    z[idx] = acc;
}

// ---------------------------------------------------------------------------
// launch
// ---------------------------------------------------------------------------
extern "C" void kernel_launch(void* const* d_in, const int* in_sizes, int n_in,
                              void* d_out, int out_size, void* d_ws, size_t ws_size,
                              hipStream_t stream) {
    const float* x          = (const float*)d_in[0];
    const float* conv_w     = (const float*)d_in[1];
    const float* conv_b     = (const float*)d_in[2];
    const float* temp_lin_w = (const float*)d_in[3];   // (256,256)
    const float* temp_lin_b = (const float*)d_in[4];   // (256)
    const float* temp_a     = (const float*)d_in[5];   // (256,1)
    const float* temp_bias  = (const float*)d_in[6];   // (100,100)
    const float* feat_lin_w = (const float*)d_in[7];   // (200,200)
    const float* feat_lin_b = (const float*)d_in[8];   // (200)
    const float* feat_a     = (const float*)d_in[9];   // (200,1)
    const float* feat_bias  = (const float*)d_in[10];  // (128,128)
    const float* emb        = (const float*)d_in[11];  // (128,8)
    const float* fc_w       = (const float*)d_in[12];  // (1,100)
    const float* fc_b       = (const float*)d_in[13];  // (1)
    float* out = (float*)d_out;

    float* ws = (float*)d_ws;
    float* xpe     = ws;                      // 409600
    float* y1      = xpe     + 409600;        // 409600  (B,W,F)
    float* left_t  = y1      + 409600;        // 819200  (B*W, 256)
    float* right_t = left_t  + 819200;        // 819200
    float* e_t     = right_t + 819200;        // 320000  (B,W,W)
    float* t_out   = e_t     + 320000;        // 409600  (B,W,F)
    float* left_f  = t_out   + 409600;        // 819200  (B,F,200)
    float* right_f = left_f  + 819200;        // 819200
    float* e_f     = right_f + 819200;        // 524288  (B,F,F)
    float* f_out   = e_f     + 524288;        // 409600  (B,F,W)
    float* cosm    = f_out   + 409600;        // 16384
    float* zbuf    = cosm    + 16384;         // 4096

    const int NTOT = B_ * W_ * F_;

    // 1) positional encoding
    pe_kernel<<<(NTOT + 255) / 256, 256, 0, stream>>>(x, xpe, NTOT);

    // 2) conv + relu  (1600 tiles, 8 waves/block -> 200 blocks)
    conv_wmma_kernel<<<200, 256, 0, stream>>>(xpe, conv_w, conv_b, y1);

    // 3a) temporal projections: (3200,128) @ (128,256)
    //     left uses temp_lin_w[:, :128], right uses temp_lin_w[:, 128:]
    gemm_wmma_kernel<<<dim3(400, 1), 256, 0, stream>>>(
        y1, temp_lin_w, nullptr, left_t,
        B_ * W_, 2 * F_, F_,
        0, F_, 1,          // A: rows (b,t), csA=1
        0, 1, 2 * F_,      // B(k,n) = temp_lin_w[n*256 + k]
        0, 2 * F_, 0, 0);
    gemm_wmma_kernel<<<dim3(400, 1), 256, 0, stream>>>(
        y1, temp_lin_w + F_, nullptr, right_t,
        B_ * W_, 2 * F_, F_,
        0, F_, 1,
        0, 1, 2 * F_,      // B(k,n) = temp_lin_w[n*256 + 128 + k]
        0, 2 * F_, 0, 0);

    // 3b) temporal edge scores + softmax
    gat_edge_kernel<<<B_ * W_, 128, 0, stream>>>(
        left_t, right_t, temp_a, temp_lin_b, temp_bias, e_t, W_, W_, 2 * F_);
    softmax_rows_kernel<<<B_ * W_, 128, 0, stream>>>(e_t, W_);

    // 3c) t_out = tanh(att @ y1), batched over B: (100x100)@(100x128)
    gemm_wmma_kernel<<<dim3(7, B_), 256, 0, stream>>>(
        e_t, y1, nullptr, t_out,
        W_, F_, W_,
        (long)W_ * W_, W_, 1,
        (long)W_ * F_, F_, 1,
        (long)W_ * F_, F_, 0, 2);

    // 4a) feature projections on v_f[b,p,w] = t_out[b,w,p]: (128,100)@(100,200)
    gemm_wmma_kernel<<<dim3(13, B_), 256, 0, stream>>>(
        t_out, feat_lin_w, nullptr, left_f,
        F_, 2 * W_, W_,
        (long)W_ * F_, 1, F_,   // A(m=p,k=w) = t_out[b, w, p]  (transposed view)
        0, 1, 2 * W_,           // B(k,n) = feat_lin_w[n*200 + k]
        (long)F_ * 2 * W_, 2 * W_, 0, 0);
    gemm_wmma_kernel<<<dim3(13, B_), 256, 0, stream>>>(
        t_out, feat_lin_w + W_, nullptr, right_f,
        F_, 2 * W_, W_,
        (long)W_ * F_, 1, F_,
        0, 1, 2 * W_,
        (long)F_ * 2 * W_, 2 * W_, 0, 0);

    // 4b) feature edge scores + softmax
    gat_edge_kernel<<<B_ * F_, 128, 0, stream>>>(
        left_f, right_f, feat_a, feat_lin_b, feat_bias, e_f, F_, F_, 2 * W_);
    softmax_rows_kernel<<<B_ * F_, 128, 0, stream>>>(e_f, F_);

    // 4c) f_out = tanh(att_f @ v_f), batched: (128x128)@(128x100) -> (B,F,W)
    gemm_wmma_kernel<<<dim3(7, B_), 256, 0, stream>>>(
        e_f, t_out, nullptr, f_out,
        F_, W_, F_,
        (long)F_ * F_, F_, 1,
        (long)W_ * F_, 1, F_,   // B(k=j,n=w) = t_out[b, w, j]
        (long)F_ * W_, W_, 0, 2);

    // 5) cosine matrix + z reduction
    cos_kernel<<<(F_ * F_ + 255) / 256, 256, 0, stream>>>(emb, cosm);
    z_kernel<<<(B_ * F_ + 255) / 256, 256, 0, stream>>>(f_out, fc_w, zbuf);

    // 6) pred[b,f] = tanh( sum_g cos[f,g] * z[b,g] + fc_b )  -> (B,1,F) flat
    gemm_wmma_kernel<<<dim3(2, 1), 256, 0, stream>>>(
        zbuf, cosm, fc_b, out,
        B_, F_, F_,
        0, F_, 1,          // A(m=b,k=g) = z[b*128+g]
        0, 1, F_,          // B(k=g,n=f) = cos[f*128+g] (cos is symmetric anyway)
        0, F_, 2, 2);
    (void)in_sizes; (void)n_in; (void)out_size; (void)ws_size;
}